// ScoringAttentionModule_21071109554653
// MI455X (gfx1250) — compile-verified
//
#include <hip/hip_runtime.h>

typedef __attribute__((ext_vector_type(2))) float v2f;
typedef __attribute__((ext_vector_type(8))) float v8f;

#define B_   32
#define D_   512
#define N_   1024
#define M_   1024

#define BT   128            // block output tile: BT x BT
#define KC   32             // K chunk staged in LDS
#define LPAD 16             // pad LDS rows: stride 144 dwords breaks bank conflicts
#define LSTR (BT + LPAD)

// grid: (64 tiles, 32 batches), block: 128 threads = 4 waves (wave32)
__global__ __launch_bounds__(128)
void score_gemm_max_kernel(const float* __restrict__ e1,
                           const float* __restrict__ e2,
                           float* __restrict__ partial)
{
    __shared__ float ldsA[KC * LSTR];   // A chunk: [k][n]  (n = output rows)
    __shared__ float ldsB[KC * LSTR];   // B chunk: [k][m]  (m = output cols)
    __shared__ float sRed[4];

    const int tid  = threadIdx.x;
    const int lane = tid & 31;
    const int wave = tid >> 5;
    const int r    = lane & 15;          // row/col within 16x16 tile
    const int hi   = (lane >> 4) & 1;    // K half-select per f32 WMMA layout

    const int tile = blockIdx.x;         // 0..63 -> 8x8 tiles of 128x128
    const int b    = blockIdx.y;         // batch
    const int n0   = (tile >> 3) * BT;
    const int m0   = (tile & 7) * BT;

    const int wy = wave >> 1;            // 2x2 wave grid over the 128x128 tile
    const int wx = wave & 1;
    const int nBase = wy * 64;
    const int mBase = wx * 64;

    const size_t e1b = (size_t)b * D_ * N_;
    const size_t e2b = (size_t)b * D_ * M_;

    v8f acc[4][4];
#pragma unroll
    for (int i = 0; i < 4; ++i)
#pragma unroll
        for (int j = 0; j < 4; ++j)
            acc[i][j] = (v8f){0.f, 0.f, 0.f, 0.f, 0.f, 0.f, 0.f, 0.f};

    for (int k0 = 0; k0 < D_; k0 += KC) {
        // ---- stage 32x128 chunks of e1 (n-slab) and e2 (m-slab) into LDS ----
        // 4096 floats per matrix = 1024 float4; 128 threads -> 8 float4 each
#pragma unroll
        for (int i = 0; i < 8; ++i) {
            const int idx = tid + i * 128;        // 0..1023
            const int row = idx >> 5;             // 0..31  (k within chunk)
            const int c4  = (idx & 31) << 2;      // 0..124 (col, float4 granularity)
            const float4 va = *(const float4*)(e1 + e1b + (size_t)(k0 + row) * N_ + n0 + c4);
            const float4 vb = *(const float4*)(e2 + e2b + (size_t)(k0 + row) * M_ + m0 + c4);
            *(float4*)(&ldsA[row * LSTR + c4]) = va;
            *(float4*)(&ldsB[row * LSTR + c4]) = vb;
        }
        __syncthreads();

        // ---- compute: 8 K-steps of 4, 16 WMMAs per step per wave ----
#pragma unroll
        for (int ks = 0; ks < KC; ks += 4) {
            const int k = ks + hi * 2;
            v2f af[4], bf[4];
#pragma unroll
            for (int i = 0; i < 4; ++i) {
                const int colA = nBase + i * 16 + r;
                af[i].x = ldsA[(k    ) * LSTR + colA];
                af[i].y = ldsA[(k + 1) * LSTR + colA];
                const int colB = mBase + i * 16 + r;
                bf[i].x = ldsB[(k    ) * LSTR + colB];
                bf[i].y = ldsB[(k + 1) * LSTR + colB];
            }
#pragma unroll
            for (int i = 0; i < 4; ++i)
#pragma unroll
                for (int j = 0; j < 4; ++j)
                    acc[i][j] = __builtin_amdgcn_wmma_f32_16x16x4_f32(
                        false, af[i], false, bf[j], (short)0, acc[i][j],
                        false, false);
        }
        __syncthreads();
    }

    // ---- max over this wave's 64x64 accumulator region ----
    float m = -__FLT_MAX__;
#pragma unroll
    for (int i = 0; i < 4; ++i)
#pragma unroll
        for (int j = 0; j < 4; ++j)
#pragma unroll
            for (int e = 0; e < 8; ++e)
                m = fmaxf(m, acc[i][j][e]);

    // cross-lane reduce within the wave (wave32!)
#pragma unroll
    for (int off = 16; off > 0; off >>= 1)
        m = fmaxf(m, __shfl_xor(m, off, 32));

    if (lane == 0) sRed[wave] = m;
    __syncthreads();
    if (tid == 0) {
        float t = fmaxf(fmaxf(sRed[0], sRed[1]), fmaxf(sRed[2], sRed[3]));
        partial[(size_t)b * 64 + tile] = t;
    }
}

// 32 blocks (one per batch) x 64 threads: reduce 64 tile partials -> out[b]
__global__ __launch_bounds__(64)
void reduce_max_kernel(const float* __restrict__ partial, float* __restrict__ out)
{
    __shared__ float s[64];
    const int b = blockIdx.x;
    const int t = threadIdx.x;
    s[t] = partial[(size_t)b * 64 + t];
    __syncthreads();
#pragma unroll
    for (int off = 32; off > 0; off >>= 1) {
        if (t < off) s[t] = fmaxf(s[t], s[t + off]);
        __syncthreads();
    }
    if (t == 0) out[b] = s[0];
}

extern "C" void kernel_launch(void* const* d_in, const int* in_sizes, int n_in,
                              void* d_out, int out_size, void* d_ws, size_t ws_size,
                              hipStream_t stream) {
    (void)in_sizes; (void)n_in; (void)out_size; (void)ws_size;
    const float* e1 = (const float*)d_in[0];   // [B, D, N] fp32
    const float* e2 = (const float*)d_in[1];   // [B, D, M] fp32
    float* out      = (float*)d_out;           // [B, 1] fp32
    float* partial  = (float*)d_ws;            // 32*64 floats = 8 KB scratch

    dim3 grid(64, 32);                         // 8x8 tiles x 32 batches
    score_gemm_max_kernel<<<grid, 128, 0, stream>>>(e1, e2, partial);
    reduce_max_kernel<<<32, 64, 0, stream>>>(partial, out);
}